// GCN_781684048182
// MI455X (gfx1250) — compile-verified
//
#include <hip/hip_runtime.h>
#include <hip/hip_bf16.h>
#include <stdint.h>

#define N_NODES 100000
#define N_EDGES 1600000
#define NFEAT   128
#define NHID    64
#define NCLASS  40

#define WAVES_PER_BLOCK 10          // 6250 row tiles = 625 blocks * 10 waves exactly
#define APAD1 132                   // padded A row stride (floats): bank = (4m+k)%64, conflict-free
#define APAD2 68

typedef __attribute__((ext_vector_type(2))) float v2f;
typedef __attribute__((ext_vector_type(8))) float v8f;
typedef __attribute__((ext_vector_type(4))) unsigned int u32x4;
typedef __attribute__((ext_vector_type(8))) unsigned int u32x8;

// Deterministic per-element dropout mask (p=0.5), scale kept values by 2.
__device__ __forceinline__ float dropout_val(float v, uint32_t idx) {
    uint32_t x = idx ^ 0x9E3779B9u;
    x *= 0x85EBCA6Bu; x ^= x >> 13;
    x *= 0xC2B2AE35u; x ^= x >> 16;
    return (x & 1u) ? v * 2.0f : 0.0f;
}

// ---------------------------------------------------------------------------
// Tensor Data Mover: DMA a 2D f32 tensor (rows x row_elems, row stride
// stride_elems) from global memory into LDS at lds_addr. Tile == whole tensor.
// D# layout per CDNA5 ISA 8.3/8.4 (count=1, type=2 image, data_size=4B).
// ---------------------------------------------------------------------------
__device__ __forceinline__ void tdm_load_2d(uint32_t lds_addr, const void* gptr,
                                            uint32_t row_elems, uint32_t rows,
                                            uint32_t stride_elems) {
    unsigned long long ga = (unsigned long long)(uintptr_t)gptr;
    u32x4 g0;
    g0.x = 1u;                                                  // count=1, user mode
    g0.y = lds_addr;                                            // lds_addr
    g0.z = (uint32_t)ga;                                        // global_addr[31:0]
    g0.w = (uint32_t)((ga >> 32) & 0x1FFFFFFu) | (2u << 30);    // global_addr[56:32], type=2
    u32x8 g1;
    g1[0] = 2u << 16;                                           // data_size=4B; mask=0
    g1[1] = (row_elems & 0xFFFFu) << 16;                        // tensor_dim0[15:0]
    g1[2] = (row_elems >> 16) | ((rows & 0xFFFFu) << 16);       // dim0[31:16] | dim1[15:0]
    g1[3] = (rows >> 16) | ((row_elems & 0xFFFFu) << 16);       // dim1[31:16] | tile_dim0
    g1[4] = rows & 0xFFFFu;                                     // tile_dim1 (tile_dim2=0)
    g1[5] = stride_elems;                                       // tensor_dim0_stride[31:0]
    g1[6] = 0;                                                  // stride[47:32] | dim1_stride lo
    g1[7] = 0;
    asm volatile("tensor_load_to_lds %0, %1" :: "s"(g0), "s"(g1) : "memory");
}

// Async 16B copy global -> LDS (per-lane), tracked with ASYNCcnt.
__device__ __forceinline__ void async_copy16(uint32_t lds_off, const void* g) {
    asm volatile("global_load_async_to_lds_b128 %0, %1, off"
                 :: "v"(lds_off), "v"((unsigned long long)(uintptr_t)g) : "memory");
}
__device__ __forceinline__ void wait_async0() {
    asm volatile("s_wait_asynccnt 0x0" ::: "memory");
}

// ---------------------------------------------------------------------------
// GEMM1: Y[100000x64] = dropout(X[100000x128]) @ W1[128x64]
// - W1 staged to LDS once per block via TDM (tensor_load_to_lds)
// - per-wave 16x128 A tile staged via global_load_async_to_lds_b128 (coalesced)
// - f32 WMMA 16x16x4, 4 column tiles, K=128
// ---------------------------------------------------------------------------
__global__ __launch_bounds__(320) void gemm1_wmma(const float* __restrict__ X,
                                                  const float* __restrict__ W,
                                                  float* __restrict__ Y) {
    __shared__ float sW[NFEAT * NHID];                    // 32 KB
    __shared__ float sA[WAVES_PER_BLOCK * 16 * APAD1];    // ~82.5 KB

    const int lane = threadIdx.x & 31;
    const int wave = threadIdx.x >> 5;
    const int row0 = (blockIdx.x * WAVES_PER_BLOCK + wave) * 16;   // always < N_NODES

    // Stage weights via Tensor Data Mover (one wave per block issues it).
    if (wave == 0) {
        tdm_load_2d((uint32_t)(uintptr_t)&sW[0], W, NHID, NFEAT, NHID);
    }

    // Stage this wave's 16x128 A tile (contiguous 8KB in global) into padded LDS.
    float* aTile = &sA[wave * 16 * APAD1];
    const float* gA = X + (size_t)row0 * NFEAT;
    #pragma unroll
    for (int r = 0; r < 16; ++r) {   // one row = 512B = 32 lanes x 16B
        uint32_t loff = (uint32_t)(uintptr_t)&aTile[r * APAD1] + (uint32_t)lane * 16u;
        async_copy16(loff, gA + (size_t)r * NFEAT + lane * 4);
    }
    wait_async0();
    if (wave == 0) __builtin_amdgcn_s_wait_tensorcnt(0);
    __syncthreads();

    const int m  = lane & 15;               // A/C row within tile
    const int kh = (lane >> 4) << 1;        // K sub-offset: 0 or 2
    const int n  = lane & 15;               // B/C column within tile
    const uint32_t rowbase = (uint32_t)(row0 + m) * NFEAT;   // dropout index base
    const float* aRow = &aTile[m * APAD1];

    v8f acc0 = {}, acc1 = {}, acc2 = {}, acc3 = {};

    for (int k0 = 0; k0 < NFEAT; k0 += 4) {
        const int ka = k0 + kh;
        v2f a;
        a.x = dropout_val(aRow[ka],     rowbase + ka);
        a.y = dropout_val(aRow[ka + 1], rowbase + ka + 1);

        const float* wr0 = &sW[ka * NHID];
        const float* wr1 = &sW[(ka + 1) * NHID];
        v2f b0, b1, b2, b3;
        b0.x = wr0[n];      b0.y = wr1[n];
        b1.x = wr0[16 + n]; b1.y = wr1[16 + n];
        b2.x = wr0[32 + n]; b2.y = wr1[32 + n];
        b3.x = wr0[48 + n]; b3.y = wr1[48 + n];

        acc0 = __builtin_amdgcn_wmma_f32_16x16x4_f32(false, a, false, b0, (short)0, acc0, false, false);
        acc1 = __builtin_amdgcn_wmma_f32_16x16x4_f32(false, a, false, b1, (short)0, acc1, false, false);
        acc2 = __builtin_amdgcn_wmma_f32_16x16x4_f32(false, a, false, b2, (short)0, acc2, false, false);
        acc3 = __builtin_amdgcn_wmma_f32_16x16x4_f32(false, a, false, b3, (short)0, acc3, false, false);
    }

    // C/D layout: lanes 0-15 -> M = vgpr idx, lanes 16-31 -> M = vgpr idx + 8
    const int mbase = (lane >> 4) << 3;
    #pragma unroll
    for (int v = 0; v < 8; ++v) {
        const size_t r = (size_t)(row0 + mbase + v) * NHID;
        Y[r +      n] = acc0[v];
        Y[r + 16 + n] = acc1[v];
        Y[r + 32 + n] = acc2[v];
        Y[r + 48 + n] = acc3[v];
    }
}

// ---------------------------------------------------------------------------
// GEMM2: Y[100000x40] = H[100000x64] @ W2[64x40] (columns padded to 48)
// ---------------------------------------------------------------------------
__global__ __launch_bounds__(320) void gemm2_wmma(const float* __restrict__ H,
                                                  const float* __restrict__ W,
                                                  float* __restrict__ Y) {
    __shared__ float sW[NHID * NCLASS];                   // 10 KB
    __shared__ float sA[WAVES_PER_BLOCK * 16 * APAD2];    // ~42.5 KB

    const int lane = threadIdx.x & 31;
    const int wave = threadIdx.x >> 5;
    const int row0 = (blockIdx.x * WAVES_PER_BLOCK + wave) * 16;

    if (wave == 0) {
        tdm_load_2d((uint32_t)(uintptr_t)&sW[0], W, NCLASS, NHID, NCLASS);
    }

    // Stage 16x64 A tile (contiguous 4KB) -> 256 x 16B chunks, 8 issues of 32 lanes.
    float* aTile = &sA[wave * 16 * APAD2];
    const float* gA = H + (size_t)row0 * NHID;
    #pragma unroll
    for (int i = 0; i < 8; ++i) {
        const int c   = i * 32 + lane;      // chunk id: row = c>>4, col16 = c&15
        const int r   = c >> 4;
        const int c16 = c & 15;
        uint32_t loff = (uint32_t)(uintptr_t)&aTile[r * APAD2] + (uint32_t)c16 * 16u;
        async_copy16(loff, gA + (size_t)r * NHID + c16 * 4);
    }
    wait_async0();
    if (wave == 0) __builtin_amdgcn_s_wait_tensorcnt(0);
    __syncthreads();

    const int m  = lane & 15;
    const int kh = (lane >> 4) << 1;
    const int n  = lane & 15;
    const float* aRow = &aTile[m * APAD2];

    v8f acc0 = {}, acc1 = {}, acc2 = {};

    for (int k0 = 0; k0 < NHID; k0 += 4) {
        const int ka = k0 + kh;
        v2f a;
        a.x = aRow[ka];
        a.y = aRow[ka + 1];

        const float* wr0 = &sW[ka * NCLASS];
        const float* wr1 = &sW[(ka + 1) * NCLASS];
        v2f b0, b1, b2;
        b0.x = wr0[n];      b0.y = wr1[n];
        b1.x = wr0[16 + n]; b1.y = wr1[16 + n];
        b2.x = (32 + n < NCLASS) ? wr0[32 + n] : 0.0f;
        b2.y = (32 + n < NCLASS) ? wr1[32 + n] : 0.0f;

        acc0 = __builtin_amdgcn_wmma_f32_16x16x4_f32(false, a, false, b0, (short)0, acc0, false, false);
        acc1 = __builtin_amdgcn_wmma_f32_16x16x4_f32(false, a, false, b1, (short)0, acc1, false, false);
        acc2 = __builtin_amdgcn_wmma_f32_16x16x4_f32(false, a, false, b2, (short)0, acc2, false, false);
    }

    const int mbase = (lane >> 4) << 3;
    #pragma unroll
    for (int v = 0; v < 8; ++v) {
        const size_t r = (size_t)(row0 + mbase + v) * NCLASS;
        Y[r +      n] = acc0[v];
        Y[r + 16 + n] = acc1[v];
        if (32 + n < NCLASS) Y[r + 32 + n] = acc2[v];
    }
}

// ---------------------------------------------------------------------------
// Graph-structure kernels
// ---------------------------------------------------------------------------
__global__ void deg_kernel(const long long* __restrict__ dst, float* __restrict__ deg) {
    int e = blockIdx.x * blockDim.x + threadIdx.x;
    if (e < N_EDGES) unsafeAtomicAdd(&deg[(int)dst[e]], 1.0f);
}

__global__ void dis_kernel(const float* __restrict__ deg, float* __restrict__ dis) {
    int i = blockIdx.x * blockDim.x + threadIdx.x;
    if (i < N_NODES) dis[i] = rsqrtf(deg[i] + 1.0f);   // +1 accounts for self-loop
}

// agg[i][:] = dis[i]^2 * xw[i][:]   (self-loop contribution, also initializes agg)
__global__ void selfloop_kernel(const float* __restrict__ xw, const float* __restrict__ dis,
                                float* __restrict__ agg, int F) {
    long long tid = (long long)blockIdx.x * blockDim.x + threadIdx.x;
    if (tid >= (long long)N_NODES * F) return;
    int i = (int)(tid / F);
    float d = dis[i];
    agg[tid] = d * d * xw[tid];
}

// For each edge, scatter-add norm * xw[src] into agg[dst]; 4 features per thread.
__global__ void scatter_edges(const long long* __restrict__ src,
                              const long long* __restrict__ dst,
                              const float* __restrict__ dis,
                              const float* __restrict__ xw,
                              float* __restrict__ agg,
                              int F, int chunks) {
    long long tid = (long long)blockIdx.x * blockDim.x + threadIdx.x;
    if (tid >= (long long)N_EDGES * chunks) return;
    int e = (int)(tid / chunks);
    int c = (int)(tid % chunks) * 4;
    int s = (int)src[e];
    int d = (int)dst[e];
    float w = dis[s] * dis[d];
    float4 v = *(const float4*)(xw + (size_t)s * F + c);
    float* ap = agg + (size_t)d * F + c;
    unsafeAtomicAdd(ap + 0, w * v.x);
    unsafeAtomicAdd(ap + 1, w * v.y);
    unsafeAtomicAdd(ap + 2, w * v.z);
    unsafeAtomicAdd(ap + 3, w * v.w);
}

__global__ void bias_kernel(float* __restrict__ a, const float* __restrict__ b,
                            int F, int do_relu) {
    long long tid = (long long)blockIdx.x * blockDim.x + threadIdx.x;
    if (tid >= (long long)N_NODES * F) return;
    int f = (int)(tid % F);
    float v = a[tid] + b[f];
    if (do_relu) v = fmaxf(v, 0.0f);
    a[tid] = v;
}

// ---------------------------------------------------------------------------
extern "C" void kernel_launch(void* const* d_in, const int* in_sizes, int n_in,
                              void* d_out, int out_size, void* d_ws, size_t ws_size,
                              hipStream_t stream) {
    const float*      X    = (const float*)d_in[0];
    const long long*  edge = (const long long*)d_in[1];
    const float*      W1   = (const float*)d_in[2];
    const float*      b1   = (const float*)d_in[3];
    const float*      W2   = (const float*)d_in[4];
    const float*      b2   = (const float*)d_in[5];
    float*            out  = (float*)d_out;

    const long long* src = edge;             // edge_index[0]
    const long long* dst = edge + N_EDGES;   // edge_index[1]

    // Workspace layout (~68 MB)
    float* deg = (float*)d_ws;                         // N
    float* dis = deg + N_NODES;                        // N
    float* xw1 = dis + N_NODES;                        // N*64
    float* h   = xw1 + (size_t)N_NODES * NHID;         // N*64
    float* xw2 = h   + (size_t)N_NODES * NHID;         // N*40

    const int T = 256;
    const int gemmBlocks  = (N_NODES / 16) / WAVES_PER_BLOCK;   // 625 exactly
    const int gemmThreads = WAVES_PER_BLOCK * 32;               // 320

    hipMemsetAsync(deg, 0, N_NODES * sizeof(float), stream);
    deg_kernel<<<(N_EDGES + T - 1) / T, T, 0, stream>>>(dst, deg);
    dis_kernel<<<(N_NODES + T - 1) / T, T, 0, stream>>>(deg, dis);

    // Layer 1
    gemm1_wmma<<<gemmBlocks, gemmThreads, 0, stream>>>(X, W1, xw1);
    {
        long long tot = (long long)N_NODES * NHID;
        selfloop_kernel<<<(int)((tot + T - 1) / T), T, 0, stream>>>(xw1, dis, h, NHID);
        long long sc = (long long)N_EDGES * (NHID / 4);
        scatter_edges<<<(int)((sc + T - 1) / T), T, 0, stream>>>(src, dst, dis, xw1, h, NHID, NHID / 4);
        bias_kernel<<<(int)((tot + T - 1) / T), T, 0, stream>>>(h, b1, NHID, 1);
    }

    // Layer 2
    gemm2_wmma<<<gemmBlocks, gemmThreads, 0, stream>>>(h, W2, xw2);
    {
        long long tot = (long long)N_NODES * NCLASS;
        selfloop_kernel<<<(int)((tot + T - 1) / T), T, 0, stream>>>(xw2, dis, out, NCLASS);
        long long sc = (long long)N_EDGES * (NCLASS / 4);
        scatter_edges<<<(int)((sc + T - 1) / T), T, 0, stream>>>(src, dst, dis, xw2, out, NCLASS, NCLASS / 4);
        bias_kernel<<<(int)((tot + T - 1) / T), T, 0, stream>>>(out, b2, NCLASS, 0);
    }
}